// msg_passing_30863634989812
// MI455X (gfx1250) — compile-verified
//
#include <hip/hip_runtime.h>

#define N_NODES 100000
#define E_EDGES 1600000
#define DIM 128

typedef float v2f __attribute__((ext_vector_type(2)));
typedef float v8f __attribute__((ext_vector_type(8)));

// ---------------------------------------------------------------- zero
__global__ __launch_bounds__(256) void zero_kernel(float* __restrict__ p, long n) {
    long i = (long)blockIdx.x * blockDim.x + threadIdx.x;
    const long stride = (long)gridDim.x * blockDim.x;
    for (; i < n; i += stride) p[i] = 0.0f;
}

// ------------------------------------------------- weight pre-pack (fragment order)
// Rearranges W[128,128] so that the WMMA B-fragment each lane needs is one
// contiguous, lane-coalesced float2:  Wp[((kk*8+ct)*32+lane)] = {W[k][col], W[k+1][col]}
// with k = kk*4 + 2*(lane>>4), col = ct*16 + (lane&15).
__global__ __launch_bounds__(256) void pack_w(const float* __restrict__ W,
                                              float* __restrict__ Wp) {
    const int i = blockIdx.x * blockDim.x + threadIdx.x;   // float2 index, 8192 total
    if (i >= 32 * 8 * 32) return;
    const int lane = i & 31;
    const int ct   = (i >> 5) & 7;
    const int kk   = i >> 8;
    const int k    = kk * 4 + ((lane >> 4) << 1);
    const int col  = ct * 16 + (lane & 15);
    float2 v;
    v.x = W[(k + 0) * DIM + col];
    v.y = W[(k + 1) * DIM + col];
    ((float2*)Wp)[i] = v;
}

// ------------------------------------------------- edge scatter (segment-sum)
// One wave32 per edge: lane l moves float4 l of the 128-float source row and
// atomically adds it (native global_atomic_add_f32) into the destination row.
__global__ __launch_bounds__(256) void scatter_kernel(const float* __restrict__ xsrc,
                                                      const int* __restrict__ ei,
                                                      float* __restrict__ agg,
                                                      float* __restrict__ cnt) {
    const int gid  = blockIdx.x * blockDim.x + threadIdx.x;
    const int wid  = gid >> 5;
    const int lane = gid & 31;
    if (wid >= E_EDGES) return;
    const int src = ei[wid];
    const int dst = ei[E_EDGES + wid];
    const float4 v = ((const float4*)(xsrc + (long)src * DIM))[lane];
    float* a = agg + (long)dst * DIM + lane * 4;
    unsafeAtomicAdd(a + 0, v.x);
    unsafeAtomicAdd(a + 1, v.y);
    unsafeAtomicAdd(a + 2, v.z);
    unsafeAtomicAdd(a + 3, v.w);
    if (lane == 0) unsafeAtomicAdd(&cnt[dst], 1.0f);
}

// ------------------------------------------------- fused SAGE GEMM (WMMA f32)
// out[r, :] = (agg[r,:]/max(cnt[r],1)) @ Wm + xdst[r,:] @ Ws + bias
// Block = 2 waves, 32 rows. Each wave owns a 16x128 output strip built from
// 8 column tiles of v_wmma_f32_16x16x4_f32 (K stepped by 4, both GEMMs fused
// into one accumulator). B fragments come pre-packed (one b64 load each).
// Optional fused row-wise L2 normalization (layer 1).
__global__ __launch_bounds__(64) void sage_gemm(const float* __restrict__ agg,
                                                const float* __restrict__ cnt,
                                                const float* __restrict__ xdst,
                                                const float* __restrict__ Wpm,
                                                const float* __restrict__ Wps,
                                                const float* __restrict__ bias,
                                                float* __restrict__ out,
                                                int nrows, int do_l2) {
    __shared__ float lA[32 * 132];   // normalized agg tile (pad 132 -> no bank conflicts)
    __shared__ float lX[32 * 132];   // x_dst tile

    const int tid  = threadIdx.x;
    const int wave = tid >> 5;
    const int lane = tid & 31;
    const int row0 = blockIdx.x * 32;

    // Stage 32 rows of agg (degree-normalized) and x into LDS, coalesced float4.
    for (int i = tid; i < 32 * 32; i += 64) {
        const int r   = i >> 5;
        const int c4  = i & 31;
        const int row = row0 + r;
        float4 a = make_float4(0.f, 0.f, 0.f, 0.f);
        float4 x = make_float4(0.f, 0.f, 0.f, 0.f);
        float inv = 0.f;
        if (row < nrows) {
            a   = ((const float4*)(agg  + (long)row * DIM))[c4];
            x   = ((const float4*)(xdst + (long)row * DIM))[c4];
            inv = 1.0f / fmaxf(cnt[row], 1.0f);
        }
        float4 an = make_float4(a.x * inv, a.y * inv, a.z * inv, a.w * inv);
        *(float4*)&lA[r * 132 + c4 * 4] = an;
        *(float4*)&lX[r * 132 + c4 * 4] = x;
    }
    __syncthreads();

    v8f acc[8];
#pragma unroll
    for (int ct = 0; ct < 8; ++ct)
#pragma unroll
        for (int r = 0; r < 8; ++r) acc[ct][r] = 0.0f;

    const int m  = lane & 15;          // row within the 16-row strip
    const int n  = lane & 15;          // column within a 16-col tile
    const int kh = (lane >> 4) << 1;   // K sub-offset: lanes 0-15 -> {0,1}, 16-31 -> {2,3}
    const float* lArow = &lA[(wave * 16 + m) * 132];
    const float* lXrow = &lX[(wave * 16 + m) * 132];
    const v2f* __restrict__ Bm = (const v2f*)Wpm;
    const v2f* __restrict__ Bs = (const v2f*)Wps;

#pragma unroll 4
    for (int kk = 0; kk < 32; ++kk) {      // K = 128 in chunks of 4
        const int k = kk * 4 + kh;
        v2f af; af.x = lArow[k]; af.y = lArow[k + 1];
        v2f xf; xf.x = lXrow[k]; xf.y = lXrow[k + 1];
#pragma unroll
        for (int ct = 0; ct < 8; ++ct) {
            const int fi = (kk * 8 + ct) * 32 + lane;   // packed fragment index
            const v2f bm = Bm[fi];
            const v2f bs = Bs[fi];
            acc[ct] = __builtin_amdgcn_wmma_f32_16x16x4_f32(
                false, af, false, bm, (short)0, acc[ct], false, false);
            acc[ct] = __builtin_amdgcn_wmma_f32_16x16x4_f32(
                false, xf, false, bs, (short)0, acc[ct], false, false);
        }
    }

    // Bias (each lane's v8f shares one column).
#pragma unroll
    for (int ct = 0; ct < 8; ++ct) {
        const float bv = bias[ct * 16 + n];
#pragma unroll
        for (int r = 0; r < 8; ++r) acc[ct][r] += bv;
    }

    // Optional fused row-wise L2 normalize: D-tile rows live in 16-lane halves,
    // so a 4-step xor shuffle reduces each row's sum of squares in-register.
    if (do_l2) {
#pragma unroll
        for (int r = 0; r < 8; ++r) {
            float ss = 0.0f;
#pragma unroll
            for (int ct = 0; ct < 8; ++ct) ss += acc[ct][r] * acc[ct][r];
#pragma unroll
            for (int off = 8; off > 0; off >>= 1) ss += __shfl_xor(ss, off, 32);
            const float sc = 1.0f / fmaxf(sqrtf(ss), 1e-12f);
#pragma unroll
            for (int ct = 0; ct < 8; ++ct) acc[ct][r] *= sc;
        }
    }

    const int mrow0 = row0 + wave * 16 + ((lane >> 4) << 3);
#pragma unroll
    for (int ct = 0; ct < 8; ++ct) {
        const int col = ct * 16 + n;
#pragma unroll
        for (int r = 0; r < 8; ++r) {
            const int row = mrow0 + r;
            if (row < nrows) out[(long)row * DIM + col] = acc[ct][r];
        }
    }
}

// ------------------------------------------------- BN batch statistics
#define BN_ROWS_PER_BLOCK 512
__global__ __launch_bounds__(128) void bn_stats(const float* __restrict__ pre, int nrows,
                                                float* __restrict__ sums,
                                                float* __restrict__ sqs) {
    const int c  = threadIdx.x;                 // one thread per column
    const int r0 = blockIdx.x * BN_ROWS_PER_BLOCK;
    int r1 = r0 + BN_ROWS_PER_BLOCK;
    if (r1 > nrows) r1 = nrows;
    float s = 0.0f, q = 0.0f;
    for (int r = r0; r < r1; ++r) {
        const float v = pre[(long)r * DIM + c];
        s += v;
        q += v * v;
    }
    unsafeAtomicAdd(&sums[c], s);
    unsafeAtomicAdd(&sqs[c], q);
}

// ------------------------------------------------- BN + LeakyReLU + L2 (wave per row)
__global__ __launch_bounds__(256) void bn_apply_kernel(const float* __restrict__ pre,
                                                       const float* __restrict__ sums,
                                                       const float* __restrict__ sqs,
                                                       const float* __restrict__ gamma,
                                                       const float* __restrict__ beta,
                                                       int nrows, float* __restrict__ outp) {
    const int wave = threadIdx.x >> 5;
    const int lane = threadIdx.x & 31;
    const int row  = blockIdx.x * 8 + wave;
    if (row >= nrows) return;
    const int c = lane * 4;
    const float invn = 1.0f / (float)nrows;
    const float4 v = ((const float4*)(pre + (long)row * DIM))[lane];
    float vv[4] = {v.x, v.y, v.z, v.w};
    float y[4];
    float ss = 0.0f;
#pragma unroll
    for (int j = 0; j < 4; ++j) {
        const float mu  = sums[c + j] * invn;
        const float var = sqs[c + j] * invn - mu * mu;
        const float rs  = rsqrtf(var + 1e-5f);
        float val = (vv[j] - mu) * rs * gamma[c + j] + beta[c + j];
        val = (val >= 0.0f) ? val : 0.01f * val;
        y[j] = val;
        ss += val * val;
    }
#pragma unroll
    for (int off = 16; off > 0; off >>= 1) ss += __shfl_xor(ss, off, 32);
    const float sc = 1.0f / fmaxf(sqrtf(ss), 1e-12f);
    const float4 o = make_float4(y[0] * sc, y[1] * sc, y[2] * sc, y[3] * sc);
    ((float4*)(outp + (long)row * DIM))[lane] = o;
}

// ---------------------------------------------------------------- launch
extern "C" void kernel_launch(void* const* d_in, const int* in_sizes, int n_in,
                              void* d_out, int out_size, void* d_ws, size_t ws_size,
                              hipStream_t stream) {
    const float* x_a    = (const float*)d_in[0];
    const float* x_b    = (const float*)d_in[1];
    const int*   ei_ab  = (const int*)d_in[2];
    const int*   ei_ba  = (const int*)d_in[3];
    const float* W0ab_m = (const float*)d_in[4];
    const float* b0ab   = (const float*)d_in[5];
    const float* W0ab_s = (const float*)d_in[6];
    const float* W0ba_m = (const float*)d_in[7];
    const float* b0ba   = (const float*)d_in[8];
    const float* W0ba_s = (const float*)d_in[9];
    const float* g0a    = (const float*)d_in[10];
    const float* be0a   = (const float*)d_in[11];
    const float* g0b    = (const float*)d_in[12];
    const float* be0b   = (const float*)d_in[13];
    const float* W1ab_m = (const float*)d_in[14];
    const float* b1ab   = (const float*)d_in[15];
    const float* W1ab_s = (const float*)d_in[16];
    const float* W1ba_m = (const float*)d_in[17];
    const float* b1ba   = (const float*)d_in[18];
    const float* W1ba_s = (const float*)d_in[19];

    const long NH = (long)N_NODES * DIM;
    const long WSZ = (long)DIM * DIM;    // 16384 floats per packed weight
    float* ws    = (float*)d_ws;
    float* agg_a = ws;
    float* agg_b = ws + NH;
    float* cnt_a = ws + 2 * NH;
    float* cnt_b = cnt_a + N_NODES;
    float* stats = cnt_b + N_NODES;      // 512 floats
    float* suma  = stats;
    float* sqa   = stats + 128;
    float* sumb  = stats + 256;
    float* sqb   = stats + 384;
    float* pre_a = stats + 512;          // becomes h_a in-place
    float* pre_b = pre_a + NH;           // becomes h_b in-place
    float* wp    = pre_b + NH;           // 8 packed weight matrices
    float* wp0ab_m = wp;
    float* wp0ab_s = wp + 1 * WSZ;
    float* wp0ba_m = wp + 2 * WSZ;
    float* wp0ba_s = wp + 3 * WSZ;
    float* wp1ab_m = wp + 4 * WSZ;
    float* wp1ab_s = wp + 5 * WSZ;
    float* wp1ba_m = wp + 6 * WSZ;
    float* wp1ba_s = wp + 7 * WSZ;
    float* out_a = (float*)d_out;        // [0, NH)
    float* out_b = out_a + NH;           // [NH, 2NH)

    const int scatter_blocks = E_EDGES / 8;   // 8 wave32 per 256-thread block
    const int gemm_blocks    = N_NODES / 32;  // 3125, exact
    const int bn_blocks      = (N_NODES + BN_ROWS_PER_BLOCK - 1) / BN_ROWS_PER_BLOCK;
    const int row_blocks     = (N_NODES + 7) / 8;
    const int pack_blocks    = (32 * 8 * 32 + 255) / 256;  // 32

    // ---- weight pre-pack (independent of graph work) ----
    pack_w<<<pack_blocks, 256, 0, stream>>>(W0ab_m, wp0ab_m);
    pack_w<<<pack_blocks, 256, 0, stream>>>(W0ab_s, wp0ab_s);
    pack_w<<<pack_blocks, 256, 0, stream>>>(W0ba_m, wp0ba_m);
    pack_w<<<pack_blocks, 256, 0, stream>>>(W0ba_s, wp0ba_s);
    pack_w<<<pack_blocks, 256, 0, stream>>>(W1ab_m, wp1ab_m);
    pack_w<<<pack_blocks, 256, 0, stream>>>(W1ab_s, wp1ab_s);
    pack_w<<<pack_blocks, 256, 0, stream>>>(W1ba_m, wp1ba_m);
    pack_w<<<pack_blocks, 256, 0, stream>>>(W1ba_s, wp1ba_s);

    // ---- layer 0 ----
    zero_kernel<<<4096, 256, 0, stream>>>(agg_a, 2 * NH + 2 * N_NODES + 512);
    scatter_kernel<<<scatter_blocks, 256, 0, stream>>>(x_a, ei_ab, agg_b, cnt_b);
    scatter_kernel<<<scatter_blocks, 256, 0, stream>>>(x_b, ei_ba, agg_a, cnt_a);
    sage_gemm<<<gemm_blocks, 64, 0, stream>>>(agg_b, cnt_b, x_b, wp0ab_m, wp0ab_s, b0ab,
                                              pre_b, N_NODES, 0);
    sage_gemm<<<gemm_blocks, 64, 0, stream>>>(agg_a, cnt_a, x_a, wp0ba_m, wp0ba_s, b0ba,
                                              pre_a, N_NODES, 0);
    bn_stats<<<bn_blocks, 128, 0, stream>>>(pre_a, N_NODES, suma, sqa);
    bn_stats<<<bn_blocks, 128, 0, stream>>>(pre_b, N_NODES, sumb, sqb);
    bn_apply_kernel<<<row_blocks, 256, 0, stream>>>(pre_a, suma, sqa, g0a, be0a, N_NODES, pre_a);
    bn_apply_kernel<<<row_blocks, 256, 0, stream>>>(pre_b, sumb, sqb, g0b, be0b, N_NODES, pre_b);

    // ---- layer 1 (h = pre, in place); L2 fused into GEMM epilogue ----
    zero_kernel<<<4096, 256, 0, stream>>>(agg_a, 2 * NH + 2 * N_NODES);
    scatter_kernel<<<scatter_blocks, 256, 0, stream>>>(pre_a, ei_ab, agg_b, cnt_b);
    scatter_kernel<<<scatter_blocks, 256, 0, stream>>>(pre_b, ei_ba, agg_a, cnt_a);
    sage_gemm<<<gemm_blocks, 64, 0, stream>>>(agg_b, cnt_b, pre_b, wp1ab_m, wp1ab_s, b1ab,
                                              out_b, N_NODES, 1);
    sage_gemm<<<gemm_blocks, 64, 0, stream>>>(agg_a, cnt_a, pre_a, wp1ba_m, wp1ba_s, b1ba,
                                              out_a, N_NODES, 1);
}